// RoutingLayer_43731357008031
// MI455X (gfx1250) — compile-verified
//
#include <hip/hip_runtime.h>
#include <math.h>

typedef __attribute__((ext_vector_type(2))) float v2f;
typedef __attribute__((ext_vector_type(4))) float v4f;
typedef __attribute__((ext_vector_type(8))) float v8f;

#define D_DIM     2048
#define B_ROWS    32768
#define NEXP      64
#define NOISE_EPS 0.01f

#define NPAIR     (D_DIM / 8)   // 256 K-step pairs (8 k-values each)
#define CH_P      16            // pairs per LDS chunk (128 k-values)
#define NCHUNK    (NPAIR / CH_P)
// per pair: 2 mats * 4 tiles * 32 lanes float4 = 256 v4f = 4 KB
#define PAIR_V4   256
#define CH_V4     (CH_P * PAIR_V4)   // 4096 v4f = 64 KB

// ---------------------------------------------------------------------------
// Prologue: build packed WMMA A-fragments for w_gate / w_noise.
// wpack[((p*2 + mat)*4 + t)*32 + lane] (float4):
//   lane<16 : { w[8p  ][e], w[8p+1][e], w[8p+4][e], w[8p+5][e] }
//   lane>=16: { w[8p+2][e], w[8p+3][e], w[8p+6][e], w[8p+7][e] }
// with e = 16t + (lane&15).  (matches V_WMMA_F32_16X16X4_F32 A-layout)
// ---------------------------------------------------------------------------
__global__ __launch_bounds__(256)
void pack_w_kernel(const float* __restrict__ wg,
                   const float* __restrict__ wn,
                   v4f* __restrict__ wpack) {
    int idx  = blockIdx.x * blockDim.x + threadIdx.x;  // 0 .. 65535
    int lane = idx & 31;
    int t    = (idx >> 5) & 3;
    int mat  = (idx >> 7) & 1;
    int p    = idx >> 8;
    int e    = 16 * t + (lane & 15);
    int kb   = 8 * p + 2 * (lane >> 4);
    const float* src = mat ? wn : wg;
    v4f v;
    v.x = src[(size_t)(kb + 0) * NEXP + e];
    v.y = src[(size_t)(kb + 1) * NEXP + e];
    v.z = src[(size_t)(kb + 4) * NEXP + e];
    v.w = src[(size_t)(kb + 5) * NEXP + e];
    wpack[idx] = v;
}

// ---------------------------------------------------------------------------
// Fused routing kernel: 8 wave32 per block, 1 wave per 16-row tile.
// Weights staged through LDS in packed-fragment form.
// ---------------------------------------------------------------------------
__global__ __launch_bounds__(256)
void moe_route_kernel(const float* __restrict__ x,
                      const v4f* __restrict__ wpack,
                      const float* __restrict__ noise,
                      float* __restrict__ out) {
    __shared__ v4f ldsbuf[CH_V4];   // 64 KB: weight chunk, later logits scratch

    const int lane = threadIdx.x & 31;
    const int wid  = threadIdx.x >> 5;
    const int row0 = (blockIdx.x * 8 + wid) * 16;

    const int lmod = lane & 15;
    const int lhi  = lane >> 4;

    // x stream pointer (B-fragment: lane<16 K{0,1}, lane>=16 K{2,3})
    const float* xp = x + (size_t)(row0 + lmod) * D_DIM + 2 * lhi;

    v8f gacc[4], nacc[4];
#pragma unroll
    for (int t = 0; t < 4; ++t)
#pragma unroll
        for (int r = 0; r < 8; ++r) { gacc[t][r] = 0.0f; nacc[t][r] = 0.0f; }

    for (int c = 0; c < NCHUNK; ++c) {
        __syncthreads();   // previous chunk fully consumed
        // ---- stage 64 KB weight chunk: coalesced b128 copies
        const v4f* gsrc = wpack + (size_t)c * CH_V4;
#pragma unroll
        for (int i = 0; i < CH_V4 / 256; ++i)
            ldsbuf[threadIdx.x + i * 256] = gsrc[threadIdx.x + i * 256];
        __syncthreads();

        const float* xc = xp + c * (CH_P * 8);
        v2f cx0 = *(const v2f*)(xc);
        v2f cx1 = *(const v2f*)(xc + 4);

        for (int p = 0; p < CH_P; ++p) {
            // prefetch next pair's x fragments one iteration ahead
            const int np = (p < CH_P - 1) ? p + 1 : p;
            v2f nx0 = *(const v2f*)(xc + 8 * np);
            v2f nx1 = *(const v2f*)(xc + 8 * np + 4);

            const v4f* wrow = ldsbuf + p * PAIR_V4;
            v4f wg4[4], wn4[4];
#pragma unroll
            for (int t = 0; t < 4; ++t) {
                wg4[t] = wrow[t * 32 + lane];          // gate frags (mat 0)
                wn4[t] = wrow[128 + t * 32 + lane];    // noise frags (mat 1)
            }
            // interleave so same-accumulator WMMAs are 8 instructions apart
#pragma unroll
            for (int t = 0; t < 4; ++t)
                gacc[t] = __builtin_amdgcn_wmma_f32_16x16x4_f32(
                    false, __builtin_shufflevector(wg4[t], wg4[t], 0, 1),
                    false, cx0, (short)0, gacc[t], false, false);
#pragma unroll
            for (int t = 0; t < 4; ++t)
                nacc[t] = __builtin_amdgcn_wmma_f32_16x16x4_f32(
                    false, __builtin_shufflevector(wn4[t], wn4[t], 0, 1),
                    false, cx0, (short)0, nacc[t], false, false);
#pragma unroll
            for (int t = 0; t < 4; ++t)
                gacc[t] = __builtin_amdgcn_wmma_f32_16x16x4_f32(
                    false, __builtin_shufflevector(wg4[t], wg4[t], 2, 3),
                    false, cx1, (short)0, gacc[t], false, false);
#pragma unroll
            for (int t = 0; t < 4; ++t)
                nacc[t] = __builtin_amdgcn_wmma_f32_16x16x4_f32(
                    false, __builtin_shufflevector(wn4[t], wn4[t], 2, 3),
                    false, cx1, (short)0, nacc[t], false, false);

            cx0 = nx0;
            cx1 = nx1;
        }
    }

    __syncthreads();   // done with weight staging; reuse LDS for logits

    // ---- Epilogue 1: fuse noise*softplus in accumulator layout -> LDS.
    // Accum elem r: lane<16 -> (expert=16t+r, row=lmod); lane>=16 -> 16t+8+r.
    float* lv = (float*)ldsbuf + (size_t)wid * 16 * NEXP;   // 8 * 4KB = 32 KB
    {
        const float* nz  = noise + (size_t)row0 * NEXP;
        const int    row = lmod;
#pragma unroll
        for (int t = 0; t < 4; ++t) {
            const int ebase = t * 16 + 8 * lhi;
#pragma unroll
            for (int h = 0; h < 2; ++h) {
                v4f nv = *(const v4f*)(nz + (size_t)row * NEXP + ebase + 4 * h);
                v4f val;
#pragma unroll
                for (int j = 0; j < 4; ++j) {
                    const int r = 4 * h + j;
                    float z  = nacc[t][r];
                    float sp = fmaxf(z, 0.0f) + log1pf(expf(-fabsf(z)));
                    val[j]   = gacc[t][r] + nv[j] * (sp + NOISE_EPS);
                }
                *(v4f*)(lv + row * NEXP + ebase + 4 * h) = val;
            }
        }
    }
    __syncthreads();

    // ---- Epilogue 2: per-row stable top-2, softmax-of-2, one-hot row write.
    if (lane < 16) {
        const int row = lane;
        float v1 = -__builtin_inff(), v2 = -__builtin_inff();
        int   i1 = 0, i2 = 0;
        for (int e = 0; e < NEXP; ++e) {
            float v = lv[row * NEXP + e];
            if (v > v1)      { v2 = v1; i2 = i1; v1 = v; i1 = e; }
            else if (v > v2) { v2 = v;  i2 = e; }
        }
        float tt = expf(v2 - v1);           // v2 <= v1 -> stable
        float g1 = 1.0f / (1.0f + tt);
        float g2 = tt   / (1.0f + tt);

        float* op = out + (size_t)(row0 + row) * NEXP;
#pragma unroll
        for (int q = 0; q < NEXP / 4; ++q) {
            v4f o;
#pragma unroll
            for (int j = 0; j < 4; ++j) {
                const int e = 4 * q + j;
                o[j] = (e == i1) ? g1 : ((e == i2) ? g2 : 0.0f);
            }
            *(v4f*)(op + 4 * q) = o;
        }
    }
}

// ---------------------------------------------------------------------------
extern "C" void kernel_launch(void* const* d_in, const int* in_sizes, int n_in,
                              void* d_out, int out_size, void* d_ws, size_t ws_size,
                              hipStream_t stream) {
    const float* x     = (const float*)d_in[0];
    const float* wg    = (const float*)d_in[1];
    const float* wn    = (const float*)d_in[2];
    const float* noise = (const float*)d_in[3];
    float* out  = (float*)d_out;
    v4f* wpack  = (v4f*)d_ws;   // 65536 float4 = 1 MB

    pack_w_kernel<<<(2 * NPAIR * 4 * 32) / 256, 256, 0, stream>>>(wg, wn, wpack);

    // 32768 rows / (8 waves * 16 rows) = 256 blocks
    moe_route_kernel<<<B_ROWS / 128, 256, 0, stream>>>(x, wpack, noise, out);
}